// QLoss_3_74689481277494
// MI455X (gfx1250) — compile-verified
//
#include <hip/hip_runtime.h>

// Q-learning TD-error kernel for MI455X (gfx1250).
// Memory-bound streaming problem (~314 MB of traffic -> ~13.5us floor at
// 23.3 TB/s). No matmul => no WMMA; we exercise the CDNA5 async
// global->LDS data-mover path (GLOBAL_LOAD_ASYNC_TO_LDS_B32 + s_wait_asynccnt).
// v2: logf -> __logf (v_log_f32, ~1 ULP) so VALU stays well under the
// bandwidth floor (accurate OCML logf was ~25 VALU/element ~= the memory
// roofline; hardware log is ~3 VALU/element and co-executes as a TRANS op).

#define NH     3
#define BDIM   2048
#define TDIM   1024
#define ADIM   7
#define GAMMA  0.99f
#define TPB    256                 // 8 wave32 per block
#define CHUNKS (TDIM / TPB)        // 4 t-chunks per (h,b) row
#define NSTAGE (TPB + 1)           // stage one extra t for next_q[t+1]

__global__ __launch_bounds__(TPB) void qloss_kernel(
    const float* __restrict__ outputs,   // [NH][B][T][A]
    const int*   __restrict__ labels,    // [B][T][3]
    const float* __restrict__ mask,      // [B][T][A]
    const float* __restrict__ reward,    // [B][T]
    float*       __restrict__ out)       // [2][NH][B][T-1]
{
    __shared__ float s_out [NSTAGE * ADIM];
    __shared__ float s_mask[NSTAGE * ADIM];
    __shared__ float s_cq  [NSTAGE];
    __shared__ float s_nq  [NSTAGE];

    // Decode block id with h fastest so the 3 heads sharing mask/labels/reward
    // chunks run back-to-back and reuse L2 for those streams.
    int bid   = blockIdx.x;
    int h     = bid % NH;
    int rest  = bid / NH;
    int chunk = rest % CHUNKS;
    int b     = rest / CHUNKS;
    int t0    = chunk * TPB;

    int n  = TDIM - t0; if (n > NSTAGE) n = NSTAGE;   // staged t-count (257 or 256)
    int n7 = n * ADIM;

    const float* gout  = outputs + ((size_t)((size_t)h * BDIM + b) * TDIM + t0) * ADIM;
    const float* gmask = mask    + ((size_t)b * TDIM + t0) * ADIM;

    // ---- Phase 1: async DMA global -> LDS (CDNA5 path, ASYNCcnt tracked) ----
    for (int i = threadIdx.x; i < n7; i += TPB) {
        // Low 32 bits of a flat __shared__ pointer == LDS byte offset.
        unsigned lds_o = (unsigned)(unsigned long long)(&s_out[i]);
        unsigned lds_m = (unsigned)(unsigned long long)(&s_mask[i]);
        unsigned off   = (unsigned)(i * 4);
        asm volatile("global_load_async_to_lds_b32 %0, %1, %2"
                     :: "v"(lds_o), "v"(off), "s"(gout) : "memory");
        asm volatile("global_load_async_to_lds_b32 %0, %1, %2"
                     :: "v"(lds_m), "v"(off), "s"(gmask) : "memory");
    }
    asm volatile("s_wait_asynccnt 0" ::: "memory");
    __syncthreads();

    // ---- Phase 2: per-t q computation: gather (current_q) + max (next_q) ----
    for (int j = threadIdx.x; j < n; j += TPB) {
        int t    = t0 + j;
        int aidx = labels[((size_t)b * TDIM + t) * 3 + h];
        float nq = -3.402823466e+38f;
        float cq = 0.0f;
        #pragma unroll
        for (int a = 0; a < ADIM; ++a) {
            // __logf -> v_log_f32 * ln2 : ~1 ULP, inputs in [0.1, 1.0]
            float qv = __logf(s_out[j * ADIM + a]) * (1.0f - s_mask[j * ADIM + a]);
            nq = fmaxf(nq, qv);
            cq = (a == aidx) ? qv : cq;
        }
        s_nq[j] = nq;
        s_cq[j] = cq;
    }
    __syncthreads();

    // ---- Phase 3: Bellman combine along T, write both output planes ----
    int n_out = (TDIM - 1) - t0; if (n_out > TPB) n_out = TPB;
    const size_t plane = (size_t)NH * BDIM * (TDIM - 1);
    float* o0 = out + ((size_t)h * BDIM + b) * (TDIM - 1);
    for (int j = threadIdx.x; j < n_out; j += TPB) {
        int t = t0 + j;
        float cur    = s_cq[j];
        float target = reward[(size_t)b * TDIM + t + 1] + GAMMA * s_nq[j + 1];
        float bell   = target - cur;
        float d      = -fminf(fmaxf(bell, -1.0f), 1.0f);
        o0[t]         = cur;   // stack plane 0: cur
        o0[plane + t] = d;     // stack plane 1: d_error
    }
}

extern "C" void kernel_launch(void* const* d_in, const int* in_sizes, int n_in,
                              void* d_out, int out_size, void* d_ws, size_t ws_size,
                              hipStream_t stream) {
    const float* outputs = (const float*)d_in[0];
    const int*   labels  = (const int*)d_in[1];
    const float* mask    = (const float*)d_in[2];
    const float* reward  = (const float*)d_in[3];
    float*       out     = (float*)d_out;

    int grid = NH * BDIM * CHUNKS;   // 24576 blocks of 256 threads
    qloss_kernel<<<grid, TPB, 0, stream>>>(outputs, labels, mask, reward, out);
}